// LogicConv2d_47897475285602
// MI455X (gfx1250) — compile-verified
//
#include <hip/hip_runtime.h>

// ---------------- problem constants ----------------
#define BATCH 16
#define CCH   16
#define HH    64
#define WW    64
#define KK    32          // num_kernels
#define LL    32          // leaves
#define OHH   60
#define OWW   60
#define PP    (OHH * OWW)        // 3600
#define NTOT  (BATCH * PP)       // 57600 (b,p) pairs
#define NNODES 63                // 32+16+8+4+2+1 tree nodes

typedef __attribute__((ext_vector_type(2))) float v2f;
typedef __attribute__((ext_vector_type(8))) float v8f;

// gate coefficient tables (multilinear extension of the 16 boolean gates)
__constant__ float OP_ONE_C[16] = {0,0,0,0,0,0,0,0,1,1,1,1,1,1,1,1};
__constant__ float OP_A_C[16]   = {0,0,1,1,0,0,1,1,-1,-1,0,0,-1,-1,0,0};
__constant__ float OP_B_C[16]   = {0,0,0,0,1,1,1,1,-1,-1,-1,-1,0,0,0,0};
__constant__ float OP_AB_C[16]  = {0,1,-1,0,-1,0,-2,-1,1,2,0,1,0,1,-1,0};

// ------------------------------------------------------------------
// Prep: softmax all 63*32 weight rows -> float4(c1,ca,cb,cab); also
// fold the (c,h,w) patch picks into flat gather offsets.
// ------------------------------------------------------------------
__global__ __launch_bounds__(256) void logic_prep_kernel(
    const float* __restrict__ w0, const float* __restrict__ w1,
    const float* __restrict__ w2, const float* __restrict__ w3,
    const float* __restrict__ w4, const float* __restrict__ w5,
    const int* __restrict__ pa_h, const int* __restrict__ pa_w,
    const int* __restrict__ pa_c, const int* __restrict__ pb_h,
    const int* __restrict__ pb_w, const int* __restrict__ pb_c,
    float4* __restrict__ coeff, int* __restrict__ offs)
{
    if (blockIdx.x == 0 && threadIdx.x < 2 * LL) {
        int i = threadIdx.x & (LL - 1);
        bool isB = threadIdx.x >= LL;
        int c = isB ? pb_c[i] : pa_c[i];
        int h = isB ? pb_h[i] : pa_h[i];
        int w = isB ? pb_w[i] : pa_w[i];
        offs[threadIdx.x] = (c * HH + h) * WW + w;
    }

    int gid = blockIdx.x * blockDim.x + threadIdx.x;
    if (gid >= NNODES * KK) return;
    int nd = gid / KK;
    int k  = gid - nd * KK;

    const float* w; int local;
    if      (nd < 32) { w = w0; local = nd;      }
    else if (nd < 48) { w = w1; local = nd - 32; }
    else if (nd < 56) { w = w2; local = nd - 48; }
    else if (nd < 60) { w = w3; local = nd - 56; }
    else if (nd < 62) { w = w4; local = nd - 60; }
    else              { w = w5; local = 0;       }
    const float* logits = w + (local * KK + k) * 16;

    float mx = logits[0];
    #pragma unroll
    for (int g = 1; g < 16; ++g) mx = fmaxf(mx, logits[g]);
    float e[16], sum = 0.f;
    #pragma unroll
    for (int g = 0; g < 16; ++g) { e[g] = expf(logits[g] - mx); sum += e[g]; }
    float inv = 1.f / sum;
    float c1 = 0.f, ca = 0.f, cb = 0.f, cab = 0.f;
    #pragma unroll
    for (int g = 0; g < 16; ++g) {
        float pg = e[g] * inv;
        c1  += pg * OP_ONE_C[g];
        ca  += pg * OP_A_C[g];
        cb  += pg * OP_B_C[g];
        cab += pg * OP_AB_C[g];
    }
    coeff[nd * KK + k] = make_float4(c1, ca, cb, cab);
}

// elementwise tree combine on one 8-row accumulator tile
__device__ __forceinline__ v8f combine8(v8f l, v8f r, float4 c) {
    v8f d;
    #pragma unroll
    for (int i = 0; i < 8; ++i)
        d[i] = c.x + c.y * l[i] + c.z * r[i] + c.w * l[i] * r[i];
    return d;
}

// ------------------------------------------------------------------
// Main: one wave per 16-position M-tile. Leaf level via
// v_wmma_f32_16x16x4_f32 (A = [1,a,b,ab] rows, B = coeff cols),
// tree fold via VALU, depth-first (5-deep register stack).
// ------------------------------------------------------------------
__global__ __launch_bounds__(256) void logic_conv_kernel(
    const float* __restrict__ x, const float4* __restrict__ gcoeff,
    const int* __restrict__ goffs, float* __restrict__ out)
{
    __shared__ float4 sc[NNODES * KK];   // 32256 B coefficient table
    __shared__ int    soff[2 * LL];

    for (int i = threadIdx.x; i < NNODES * KK; i += 256) sc[i] = gcoeff[i];
    if (threadIdx.x < 2 * LL) soff[threadIdx.x] = goffs[threadIdx.x];
    __syncthreads();

    const int lane = threadIdx.x & 31;
    const int wave = threadIdx.x >> 5;
    const int tile = blockIdx.x * 8 + wave;     // 0..3599, 16 (b,p) rows each
    const int m    = lane & 15;
    const bool hi  = lane >= 16;

    int t     = tile * 16 + m;
    int batch = t / PP;
    int p     = t - batch * PP;
    int oh    = p / OWW;
    int ow    = p - oh * OWW;
    int base  = batch * (CCH * HH * WW) + oh * WW + ow;

    const int k0 = lane & 15;
    constexpr int lvloff[6] = {0, 32, 48, 56, 60, 62};

    v8f stk[5][2];
    v8f cur0 = {}, cur1 = {};
    v8f zero = {};

    #pragma unroll
    for (int leaf = 0; leaf < LL; ++leaf) {
        float av = x[base + soff[leaf]];
        float bv = x[base + soff[LL + leaf]];

        // A 16x4 f32: v0 = K0 (1.0) | K2 (b);  v1 = K1 (a) | K3 (a*b)
        v2f A;
        A.x = hi ? bv        : 1.0f;
        A.y = hi ? av * bv   : av;

        // B 4x16 f32: v0 = row c1 | row cb;  v1 = row ca | row cab
        float4 cA0 = sc[leaf * KK + k0];
        float4 cA1 = sc[leaf * KK + 16 + k0];
        v2f B0, B1;
        B0.x = hi ? cA0.z : cA0.x;
        B0.y = hi ? cA0.w : cA0.y;
        B1.x = hi ? cA1.z : cA1.x;
        B1.y = hi ? cA1.w : cA1.y;

        cur0 = __builtin_amdgcn_wmma_f32_16x16x4_f32(false, A, false, B0,
                                                     (short)0, zero, false, false);
        cur1 = __builtin_amdgcn_wmma_f32_16x16x4_f32(false, A, false, B1,
                                                     (short)0, zero, false, false);

        // carry-fold: combine while trailing ones in `leaf`
        #pragma unroll
        for (int lvl = 0; lvl < 5; ++lvl) {
            const int mask = (2 << lvl) - 1;           // bits 0..lvl
            if ((leaf & mask) == mask) {               // folds at compile time
                int node = lvloff[lvl + 1] + (leaf >> (lvl + 1));
                float4 c0 = sc[node * KK + k0];
                float4 c1 = sc[node * KK + 16 + k0];
                cur0 = combine8(stk[lvl][0], cur0, c0);
                cur1 = combine8(stk[lvl][1], cur1, c1);
            }
        }
        // store as left sibling at the first 0-bit level (none when leaf==31)
        #pragma unroll
        for (int s = 0; s < 5; ++s) {
            if ((leaf & ((2 << s) - 1)) == ((1 << s) - 1)) {
                stk[s][0] = cur0;
                stk[s][1] = cur1;
            }
        }
    }

    // root: cur0 = k 0..15, cur1 = k 16..31; D rows: lanes<16 -> M=r, else M=8+r
    #pragma unroll
    for (int r = 0; r < 8; ++r) {
        int mrow = hi ? (8 + r) : r;
        int tt   = tile * 16 + mrow;
        int bb   = tt / PP;
        int ppos = tt - bb * PP;
        int ohh  = ppos / OWW;
        int oww  = ppos - ohh * OWW;
        out[((bb * KK + k0)      * OHH + ohh) * OWW + oww] = cur0[r];
        out[((bb * KK + k0 + 16) * OHH + ohh) * OWW + oww] = cur1[r];
    }
}

// ------------------------------------------------------------------
extern "C" void kernel_launch(void* const* d_in, const int* in_sizes, int n_in,
                              void* d_out, int out_size, void* d_ws, size_t ws_size,
                              hipStream_t stream) {
    const float* x  = (const float*)d_in[0];
    const float* w0 = (const float*)d_in[1];
    const float* w1 = (const float*)d_in[2];
    const float* w2 = (const float*)d_in[3];
    const float* w3 = (const float*)d_in[4];
    const float* w4 = (const float*)d_in[5];
    const float* w5 = (const float*)d_in[6];
    const int* pa_h = (const int*)d_in[7];
    const int* pa_w = (const int*)d_in[8];
    const int* pa_c = (const int*)d_in[9];
    const int* pb_h = (const int*)d_in[10];
    const int* pb_w = (const int*)d_in[11];
    const int* pb_c = (const int*)d_in[12];

    float4* coeff = (float4*)d_ws;
    int*    offs  = (int*)((char*)d_ws + sizeof(float4) * NNODES * KK);

    hipLaunchKernelGGL(logic_prep_kernel, dim3(8), dim3(256), 0, stream,
                       w0, w1, w2, w3, w4, w5,
                       pa_h, pa_w, pa_c, pb_h, pb_w, pb_c, coeff, offs);

    hipLaunchKernelGGL(logic_conv_kernel, dim3(NTOT / 16 / 8), dim3(256), 0, stream,
                       x, (const float4*)coeff, (const int*)offs, (float*)d_out);
}